// TensorProduct_15702400434755
// MI455X (gfx1250) — compile-verified
//
#include <hip/hip_runtime.h>
#include <stdint.h>

#define ROW_LEN 256
#define MULC 64
#define ROWS_PER_BLOCK 16
#define THREADS (ROWS_PER_BLOCK * 32)

typedef uint32_t u32x4 __attribute__((ext_vector_type(4)));
typedef uint32_t u32x8 __attribute__((ext_vector_type(8)));

// Issue a TDM DMA: copy a 2D tile (dim0 f32 wide, `rows` rows, row stride =
// dim0) from global memory into LDS at byte offset lds_byte. Descriptor
// packing per CDNA5 ISA 8.3 (group 0) / 8.4 (group 1). 2D tensor -> VADDR2/3
// are NULL, so only two SGPR groups are supplied.
__device__ __forceinline__ void tdm_load_tile_2d(uint32_t lds_byte,
                                                 const void* gptr,
                                                 uint32_t dim0,
                                                 uint32_t rows) {
  const uint64_t ga = (uint64_t)(uintptr_t)gptr;

  u32x4 g0;
  g0[0] = 1u;                                   // [1:0] count=1 (valid D#), rest 0
  g0[1] = lds_byte;                             // [63:32] lds_addr (bytes)
  g0[2] = (uint32_t)ga;                         // [95:64] global_addr lo
  g0[3] = (uint32_t)((ga >> 32) & 0x01FFFFFFull)// [120:96] global_addr hi (57-bit)
        | (2u << 30);                           // [127:126] type=2 ("image")

  u32x8 g1;
  g1[0] = 0x00020000u;                          // data_size=2 -> 4 bytes; mask/flags/pad=0
  g1[1] = (dim0 & 0xFFFFu) << 16;               // [63:48] tensor_dim0[15:0]
  g1[2] = (dim0 >> 16) | ((rows & 0xFFFFu) << 16); // dim0 hi | tensor_dim1 lo
  g1[3] = (rows >> 16) | ((dim0 & 0xFFFFu) << 16); // dim1 hi | tile_dim0
  g1[4] = rows & 0xFFFFu;                       // tile_dim1=rows, tile_dim2=0
  g1[5] = dim0;                                 // tensor_dim0_stride lo (48-bit)
  g1[6] = 0u;                                   // stride0 hi | stride1 lo
  g1[7] = 0u;                                   // stride1 hi

  asm volatile("tensor_load_to_lds %0, %1" :: "s"(g0), "s"(g1) : "memory");
}

__global__ __launch_bounds__(THREADS) void tp_tdm_kernel(
    const float* __restrict__ x1, const float* __restrict__ x2,
    const float* __restrict__ w, float* __restrict__ out, int nrows) {
  __shared__ float sx1[ROWS_PER_BLOCK * ROW_LEN];   // 16 KB
  __shared__ float sw [ROWS_PER_BLOCK * ROW_LEN];   // 16 KB

  const int row0 = blockIdx.x * ROWS_PER_BLOCK;
  uint32_t tile_rows = (uint32_t)min(ROWS_PER_BLOCK, nrows - row0);

  // Wave 0 of the workgroup drives the Tensor Data Mover for both tiles.
  if (threadIdx.x < 32) {
    tdm_load_tile_2d((uint32_t)(uintptr_t)&sx1[0],
                     x1 + (size_t)row0 * ROW_LEN, ROW_LEN, tile_rows);
    tdm_load_tile_2d((uint32_t)(uintptr_t)&sw[0],
                     w + (size_t)row0 * ROW_LEN, ROW_LEN, tile_rows);
    __builtin_amdgcn_s_wait_tensorcnt(0);
  }
  __syncthreads();

  const int wv = threadIdx.x >> 5;     // wave id within block = row within tile
  const int ln = threadIdx.x & 31;     // lane
  const int z  = row0 + wv;
  if (z >= nrows) return;

  const float* __restrict__ lx1 = sx1 + wv * ROW_LEN;
  const float* __restrict__ lw  = sw  + wv * ROW_LEN;

  // x2 row: 16 B, uniform across the wave.
  const float4 xv = ((const float4*)x2)[z];
  const float x20 = xv.x, x21 = xv.y, x22 = xv.z, x23 = xv.w;

  float* __restrict__ orow = out + (size_t)z * ROW_LEN;
  const float INV3 = 0.57735026918962576f;   // 1/sqrt(3)

#pragma unroll
  for (int t = 0; t < 2; ++t) {
    const int u = ln + t * 32;

    const float x10 = lx1[u];                  // stride-1: conflict-free
    const float a   = lx1[MULC + 3 * u + 0];   // stride-3: conflict-free (gcd(3,64)=1)
    const float b   = lx1[MULC + 3 * u + 1];
    const float c   = lx1[MULC + 3 * u + 2];

    const float w0 = lw[u];
    const float w1 = lw[MULC + u];
    const float w2 = lw[2 * MULC + u];
    const float w3 = lw[3 * MULC + u];

    // out_0e[u] = w0*x1_0*x2_0 + w3*dot(x1_1[u,:], x2_1)*inv_sqrt3
    const float dot = a * x21 + b * x22 + c * x23;
    orow[u] = w0 * x10 * x20 + w3 * dot * INV3;

    // out_1o[u,j] = (w1*x1_0*x2_1[j] + w2*x2_0*x1_1[u,j]) * inv_sqrt3
    const float p1 = w1 * x10 * INV3;
    const float p2 = w2 * x20 * INV3;
    orow[MULC + 3 * u + 0] = p1 * x21 + p2 * a;
    orow[MULC + 3 * u + 1] = p1 * x22 + p2 * b;
    orow[MULC + 3 * u + 2] = p1 * x23 + p2 * c;
  }
}

extern "C" void kernel_launch(void* const* d_in, const int* in_sizes, int n_in,
                              void* d_out, int out_size, void* d_ws, size_t ws_size,
                              hipStream_t stream) {
  const float* x1 = (const float*)d_in[0];
  const float* x2 = (const float*)d_in[1];
  const float* w  = (const float*)d_in[2];
  float* out      = (float*)d_out;

  const int nrows = in_sizes[1] / 4;   // x2 is (Z, 4)
  const int blocks = (nrows + ROWS_PER_BLOCK - 1) / ROWS_PER_BLOCK;

  tp_tdm_kernel<<<blocks, THREADS, 0, stream>>>(x1, x2, w, out, nrows);
}